// SRALNet_29850022707395
// MI455X (gfx1250) — compile-verified
//
#include <hip/hip_runtime.h>
#include <math.h>

// ---------------------------------------------------------------------------
// SRALNet forward for MI455X (gfx1250, wave32, WMMA).
//   x:        (16, 128, 30, 40) f32   -> treated as (N=16, C=128, P=1200)
//   centroids:(64, 5, 128) f32        -> c_flat (M=320, C=128)
//   out:      (16, 8192) f32
// ---------------------------------------------------------------------------

#define NBATCH 16
#define CDIM   128
#define PPTS   1200
#define KCLU   64
#define S1     5
#define MROWS  (KCLU * S1)   // 320
#define ALPHA  100.0f
#define EPSV   1e-12f

typedef __attribute__((ext_vector_type(2))) float v2f;
typedef __attribute__((ext_vector_type(8))) float v8f;

// workspace layout (float offsets)
#define WS_W     0
#define WS_B     (WS_W + MROWS * CDIM)                 // 40960
#define WS_MULT  (WS_B + MROWS)                        // 41280
#define WS_SSUM  (WS_MULT + NBATCH * KCLU * PPTS)      // +1228800
#define WS_RN    (WS_SSUM + NBATCH * KCLU)             // +1024
// total floats = WS_RN + 1024  ~= 1,272,128 floats ~= 5.1 MB

// ---------------------------------------------------------------------------
// Kernel 1: W = 2*alpha*c_flat ; b = -alpha * ||c_flat_row||
// ---------------------------------------------------------------------------
__global__ __launch_bounds__(128) void prep_kernel(const float* __restrict__ cent,
                                                   float* __restrict__ ws) {
  __shared__ float red[128];
  const int r = blockIdx.x;    // 0..319
  const int t = threadIdx.x;   // 0..127
  const float v = cent[r * CDIM + t];
  ws[WS_W + r * CDIM + t] = 2.0f * ALPHA * v;
  red[t] = v * v;
  __syncthreads();
  for (int off = 64; off > 0; off >>= 1) {
    if (t < off) red[t] += red[t + off];
    __syncthreads();
  }
  if (t == 0) ws[WS_B + r] = -ALPHA * sqrtf(red[0]);
}

// ---------------------------------------------------------------------------
// Kernel 2: per (n, 16-column P-tile): conv = W*X + b via f32 WMMA, then
// coupled double softmax + mult, writing mult (N,K,P) to workspace.
// One wave (32 threads) per block.  conv_out never leaves LDS.
// ---------------------------------------------------------------------------
__global__ __launch_bounds__(32) void gemm1_softmax_kernel(const float* __restrict__ x,
                                                           float* __restrict__ ws) {
  __shared__ float Xs[CDIM][16];    // X tile: 128 rows (C) x 16 cols (P)  (8 KB)
  __shared__ float Cv[MROWS][16];   // conv tile: 320 x 16                 (20 KB)

  const int pt   = blockIdx.x;      // 0..74
  const int n    = blockIdx.y;      // 0..15
  const int lane = threadIdx.x;     // 0..31
  const int p0   = pt * 16;
  const int col  = lane & 15;
  const int half = lane >> 4;       // 0 or 1

  // stage X tile into LDS (half-wave per row parity, coalesced 16-wide rows)
  const float* xb = x + (size_t)n * CDIM * PPTS;
  for (int c = half; c < CDIM; c += 2)
    Xs[c][col] = xb[(size_t)c * PPTS + p0 + col];
  __syncthreads();

  const float* W  = ws + WS_W;
  const float* Bv = ws + WS_B;
  const int koff  = half ? 2 : 0;   // A/B K-offset per half-wave (ISA layout)

  for (int mt = 0; mt < MROWS / 16; ++mt) {
    v8f acc = {};
    const int arow = mt * 16 + col;             // A: M = lane&15
    const float* wrow = W + (size_t)arow * CDIM + koff;
    for (int k = 0; k < CDIM; k += 4) {
      // A 16x4 f32: vgpr0 = K=k+koff, vgpr1 = K=k+koff+1 (8B aligned pair)
      v2f a = *(const v2f*)(wrow + k);
      // B 4x16 f32: vgpr v -> row K=k+koff+v, N = lane&15
      v2f b;
      b[0] = Xs[k + koff][col];
      b[1] = Xs[k + koff + 1][col];
      acc = __builtin_amdgcn_wmma_f32_16x16x4_f32(
          /*neg_a=*/false, a, /*neg_b=*/false, b,
          /*c_mod=*/(short)0, acc, /*reuse_a=*/false, /*reuse_b=*/false);
    }
    // C/D layout: vgpr r -> M = mt*16 + r (+8 for lanes 16-31), N = lane&15
    const int mbase = mt * 16 + (half ? 8 : 0);
    for (int r = 0; r < 8; ++r)
      Cv[mbase + r][col] = acc[r] + Bv[mbase + r];
  }
  __syncthreads();

  // softmax over K (per s) x softmax over S (per k); mult = prod_s(1+sa)
  if (lane < 16) {
    const int p = lane;
    float mA[S1], sA[S1];
    for (int s = 0; s < S1; ++s) { mA[s] = -3.0e38f; sA[s] = 0.0f; }
    for (int k = 0; k < KCLU; ++k) {
      for (int s = 0; s < S1; ++s) {
        const float v = Cv[k * S1 + s][p];
        const float mn = fmaxf(mA[s], v);
        sA[s] = sA[s] * __expf(mA[s] - mn) + __expf(v - mn);
        mA[s] = mn;
      }
    }
    float* MU = ws + WS_MULT + (size_t)n * KCLU * PPTS;
    for (int k = 0; k < KCLU; ++k) {
      float vv[S1];
      float mB = -3.0e38f;
      for (int s = 0; s < S1; ++s) { vv[s] = Cv[k * S1 + s][p]; mB = fmaxf(mB, vv[s]); }
      float sB = 0.0f;
      for (int s = 0; s < S1; ++s) sB += __expf(vv[s] - mB);
      float mult = 1.0f;
      for (int s = 0; s < S1; ++s) {
        const float sa = (__expf(vv[s] - mA[s]) / sA[s]) * (__expf(vv[s] - mB) / sB);
        mult *= (1.0f + sa);
      }
      MU[(size_t)k * PPTS + p0 + p] = mult;
    }
  }
}

// ---------------------------------------------------------------------------
// Kernel 3: S[n,k] = sum_p mult[n,k,p]   (one wave32 per (n,k))
// ---------------------------------------------------------------------------
__global__ __launch_bounds__(32) void ssum_kernel(float* __restrict__ ws) {
  const int nk = blockIdx.x;  // 0..1023
  const float* MU = ws + WS_MULT + (size_t)nk * PPTS;
  float s = 0.0f;
  for (int p = threadIdx.x; p < PPTS; p += 32) s += MU[p];
  for (int off = 16; off > 0; off >>= 1) s += __shfl_down(s, off, 32);
  if (threadIdx.x == 0) ws[WS_SSUM + nk] = s;
}

// ---------------------------------------------------------------------------
// Kernel 4: out_raw[n,k,c] = sum_p mult[n,k,p]*x[n,c,p] - rep[k,c]*S[n,k]
// via f32 WMMA: M=k(16), N=c(16), K=p (300 steps of 4). One wave per tile.
// ---------------------------------------------------------------------------
__global__ __launch_bounds__(32) void gemm2_kernel(const float* __restrict__ x,
                                                   const float* __restrict__ cent,
                                                   const float* __restrict__ ws,
                                                   float* __restrict__ out) {
  const int nt   = blockIdx.x;   // 0..7  (c tile)
  const int mt   = blockIdx.y;   // 0..3  (k tile)
  const int n    = blockIdx.z;   // 0..15
  const int lane = threadIdx.x;
  const int l15  = lane & 15;
  const int half = lane >> 4;
  const int koff = half ? 2 : 0;

  // A: mult rows (k), K-dim = p.  B: x columns -> B[p][c] = x[c][p].
  const float* MU = ws + WS_MULT + ((size_t)n * KCLU + mt * 16 + l15) * PPTS + koff;
  const float* XB = x + ((size_t)n * CDIM + nt * 16 + l15) * PPTS + koff;

  v8f acc = {};
  for (int p = 0; p < PPTS; p += 4) {
    v2f a = *(const v2f*)(MU + p);
    v2f b = *(const v2f*)(XB + p);
    acc = __builtin_amdgcn_wmma_f32_16x16x4_f32(
        false, a, false, b, (short)0, acc, false, false);
  }

  const float* Ssum = ws + WS_SSUM + n * KCLU;
  const int kbase = mt * 16 + (half ? 8 : 0);
  const int cc    = nt * 16 + l15;
  for (int r = 0; r < 8; ++r) {
    const int kk = kbase + r;
    const float rep = cent[(size_t)(kk * S1) * CDIM + cc];  // centroids[k,0,c]
    out[(size_t)n * KCLU * CDIM + (size_t)kk * CDIM + cc] = acc[r] - rep * Ssum[kk];
  }
}

// ---------------------------------------------------------------------------
// Kernel 5: per (n,k) row L2 normalization; stash post-norm row sumsq.
// ---------------------------------------------------------------------------
__global__ __launch_bounds__(128) void rownorm_kernel(float* __restrict__ out,
                                                      float* __restrict__ ws) {
  __shared__ float red[128];
  const int nk = blockIdx.x;   // 0..1023
  const int t  = threadIdx.x;  // 0..127
  const float v = out[(size_t)nk * CDIM + t];
  red[t] = v * v;
  __syncthreads();
  for (int off = 64; off > 0; off >>= 1) {
    if (t < off) red[t] += red[t + off];
    __syncthreads();
  }
  const float sum   = red[0];
  const float scale = 1.0f / fmaxf(sqrtf(sum), EPSV);
  out[(size_t)nk * CDIM + t] = v * scale;
  if (t == 0) ws[WS_RN + nk] = sum * scale * scale;
}

// ---------------------------------------------------------------------------
// Kernel 6: per-n global L2 normalization over 8192 elements.
// ---------------------------------------------------------------------------
__global__ __launch_bounds__(256) void finalnorm_kernel(float* __restrict__ out,
                                                        const float* __restrict__ ws) {
  __shared__ float red[64];
  __shared__ float scale_s;
  const int n = blockIdx.x;
  const int t = threadIdx.x;
  if (t < 64) red[t] = ws[WS_RN + n * KCLU + t];
  __syncthreads();
  if (t == 0) {
    float s = 0.0f;
    for (int i = 0; i < 64; ++i) s += red[i];
    scale_s = 1.0f / fmaxf(sqrtf(s), EPSV);
  }
  __syncthreads();
  const float sc = scale_s;
  for (int i = t; i < KCLU * CDIM; i += 256)
    out[(size_t)n * KCLU * CDIM + i] *= sc;
}

// ---------------------------------------------------------------------------
extern "C" void kernel_launch(void* const* d_in, const int* in_sizes, int n_in,
                              void* d_out, int out_size, void* d_ws, size_t ws_size,
                              hipStream_t stream) {
  (void)in_sizes; (void)n_in; (void)out_size; (void)ws_size;
  const float* x    = (const float*)d_in[0];   // (16,128,30,40)
  const float* cent = (const float*)d_in[1];   // (64,5,128)
  float* out = (float*)d_out;                  // (16,8192)
  float* ws  = (float*)d_ws;

  prep_kernel<<<MROWS, 128, 0, stream>>>(cent, ws);
  gemm1_softmax_kernel<<<dim3(PPTS / 16, NBATCH), 32, 0, stream>>>(x, ws);
  ssum_kernel<<<NBATCH * KCLU, 32, 0, stream>>>(ws);
  gemm2_kernel<<<dim3(CDIM / 16, KCLU / 16, NBATCH), 32, 0, stream>>>(x, cent, ws, out);
  rownorm_kernel<<<NBATCH * KCLU, 128, 0, stream>>>(out, ws);
  finalnorm_kernel<<<NBATCH, 256, 0, stream>>>(out, ws);
}